// BiSPA_79087527788639
// MI455X (gfx1250) — compile-verified
//
#include <hip/hip_runtime.h>
#include <hip/hip_bf16.h>

// ---------------------------------------------------------------------------
// BiSPA forward for MI455X (gfx1250, wave32). All dense math goes through
// V_WMMA_F32_16X16X4_F32 (fp32 A/B, fp32 accum) to match the fp32 reference
// while using the CDNA5 matrix pipeline.
// Fragment layouts per CDNA5 ISA 7.12.2:
//   A (16x4):  lane = 16*half + r ; v0 = A[r][2*half], v1 = A[r][2*half+1]
//   B (4x16):  v0 = B[2*half][r],  v1 = B[2*half+1][r]
//   C (16x16): vgpr i: row = i + 8*half, col = r
// ---------------------------------------------------------------------------

typedef float v2f __attribute__((ext_vector_type(2)));
typedef float v8f __attribute__((ext_vector_type(8)));

#define BB 192
#define SS 192
#define EE 512
#define HH 8
#define DD 64
#define NCHUNK 6
#define MM (BB * SS)

static __device__ __forceinline__ v8f wmma4(v2f a, v2f b, v8f c) {
  return __builtin_amdgcn_wmma_f32_16x16x4_f32(
      /*neg_a=*/false, a, /*neg_b=*/false, b,
      /*c_mod=*/(short)0, c, /*reuse_a=*/false, /*reuse_b=*/false);
}

// ---------------------------------------------------------------------------
// GEMM: C[m, coff+n] = act( A[m,:K] . Wt[n,:K] + bias[n] )
//   A   : M x K  (row-major, leading dim lda)
//   Wt  : N x K  (row-major -> computes A @ W^T)
//   C   : M x ldc (written at column offset coff)
// Block = 256 threads = 8 waves in a 4(M) x 2(N) grid; block tile 128 x 128;
// K slab = 32. Each wave owns a 32(M) x 64(N) strip: 8 accumulators,
// 2 A + 4 B fragment loads feed 8 WMMAs per K=4 step.
// ---------------------------------------------------------------------------
__global__ __launch_bounds__(256) void gemm_wmma_f32(
    const float* __restrict__ A, int lda,
    const float* __restrict__ Wt,
    const float* __restrict__ bias,
    float* __restrict__ C, int ldc, int coff,
    int K, int relu) {
  __shared__ float As[128][33];
  __shared__ float Bs[128][33];

  const int t = threadIdx.x;
  const int lane = t & 31;
  const int wv = t >> 5;          // 0..7
  const int wvM = wv & 3;         // 0..3 -> 32-row strip
  const int wvN = wv >> 2;        // 0..1 -> 64-col strip
  const int half = lane >> 4;     // 0/1
  const int r = lane & 15;        // 0..15
  const int mblk = blockIdx.y * 128;
  const int nblk = blockIdx.x * 128;

  const v8f vzero = {0.f, 0.f, 0.f, 0.f, 0.f, 0.f, 0.f, 0.f};
  v8f acc[2][4];
#pragma unroll
  for (int mi = 0; mi < 2; ++mi)
#pragma unroll
    for (int nt = 0; nt < 4; ++nt) acc[mi][nt] = vzero;

  const int tx = t & 7;    // 8 x float4 covers 32 K-columns
  const int ty = t >> 3;   // 32 rows per pass

  for (int k0 = 0; k0 < K; k0 += 32) {
    __syncthreads();
    // Stage A slab: 128 x 32
#pragma unroll
    for (int p = 0; p < 4; ++p) {
      const int row = ty + p * 32;
      const float4 v =
          *(const float4*)(A + (size_t)(mblk + row) * lda + k0 + tx * 4);
      As[row][tx * 4 + 0] = v.x;
      As[row][tx * 4 + 1] = v.y;
      As[row][tx * 4 + 2] = v.z;
      As[row][tx * 4 + 3] = v.w;
    }
    // Stage B slab: 128 x 32
#pragma unroll
    for (int p = 0; p < 4; ++p) {
      const int row = ty + p * 32;
      const float4 v =
          *(const float4*)(Wt + (size_t)(nblk + row) * K + k0 + tx * 4);
      Bs[row][tx * 4 + 0] = v.x;
      Bs[row][tx * 4 + 1] = v.y;
      Bs[row][tx * 4 + 2] = v.z;
      Bs[row][tx * 4 + 3] = v.w;
    }
    // Prefetch next K slab while we compute this one.
    if (k0 + 32 < K) {
      __builtin_prefetch(A + (size_t)(mblk + ty) * lda + k0 + 32, 0, 1);
      __builtin_prefetch(Wt + (size_t)(nblk + ty) * K + k0 + 32, 0, 1);
    }
    __syncthreads();

#pragma unroll
    for (int kk = 0; kk < 32; kk += 4) {
      v2f a0, a1;
      a0.x = As[wvM * 32 + r][kk + 2 * half];
      a0.y = As[wvM * 32 + r][kk + 2 * half + 1];
      a1.x = As[wvM * 32 + 16 + r][kk + 2 * half];
      a1.y = As[wvM * 32 + 16 + r][kk + 2 * half + 1];
#pragma unroll
      for (int nt = 0; nt < 4; ++nt) {
        v2f b;
        b.x = Bs[wvN * 64 + nt * 16 + r][kk + 2 * half];
        b.y = Bs[wvN * 64 + nt * 16 + r][kk + 2 * half + 1];
        acc[0][nt] = wmma4(a0, b, acc[0][nt]);
        acc[1][nt] = wmma4(a1, b, acc[1][nt]);
      }
    }
  }

  // Epilogue: bias (+ReLU) and store.
#pragma unroll
  for (int nt = 0; nt < 4; ++nt) {
    const int col = nblk + wvN * 64 + nt * 16 + r;
    const float bv = bias[col];
#pragma unroll
    for (int mi = 0; mi < 2; ++mi) {
#pragma unroll
      for (int i = 0; i < 8; ++i) {
        const int row = mblk + wvM * 32 + mi * 16 + i + 8 * half;
        float v = acc[mi][nt][i] + bv;
        if (relu) v = fmaxf(v, 0.0f);
        C[(size_t)row * ldc + coff + col] = v;
      }
    }
  }
}

// ---------------------------------------------------------------------------
// Sliding-window attention, chunked 3-neighbor formulation.
// One 128-thread block (4 waves) per (head h, chunk n, batch b).
//   q  : 32 x 64      k3,v3 : 96 x 64 (chunks n-1,n,n+1; OOB rows zeroed)
//   scores = q.k3^T /8  -> masked softmax -> out = P.v3  (all via WMMA)
// transKV=1 swaps (b, j) indexing of K/V rows (horizontal branch).
// ---------------------------------------------------------------------------
__global__ __launch_bounds__(128) void attn_wmma_f32(
    const float* __restrict__ qkv,  // (B*S) x 1536 : [q | k | v]
    float* __restrict__ ctx,        // (B*S) x 512
    int transKV) {
  __shared__ float qs[32][65];
  __shared__ float ks[96][65];
  __shared__ float vs[96][64];
  __shared__ float ps[32][97];

  const int h = blockIdx.x;
  const int n = blockIdx.y;
  const int b = blockIdx.z;
  const int t = threadIdx.x;
  const int lane = t & 31;
  const int wv = t >> 5;        // 0..3
  const int half = lane >> 4;
  const int r = lane & 15;

  const int tx = t & 15;        // 16 x float4 covers 64 columns
  const int ty = t >> 4;        // 8 rows per pass

  // Load Q (rows of this chunk)
#pragma unroll
  for (int p = 0; p < 4; ++p) {
    const int row = ty + p * 8;
    const int s = n * 32 + row;
    const float4 v =
        *(const float4*)(qkv + ((size_t)b * SS + s) * 1536 + h * 64 + tx * 4);
    qs[row][tx * 4 + 0] = v.x;
    qs[row][tx * 4 + 1] = v.y;
    qs[row][tx * 4 + 2] = v.z;
    qs[row][tx * 4 + 3] = v.w;
  }
  // Load K3 / V3 (chunks n-1, n, n+1), transposed indexing for h-branch.
#pragma unroll
  for (int p = 0; p < 12; ++p) {
    const int row = ty + p * 8;                 // 0..95
    const int j = (n - 1) * 32 + row;           // absolute key position
    const bool ok = (j >= 0) && (j < SS);
    float4 kv = make_float4(0.f, 0.f, 0.f, 0.f);
    float4 vv = make_float4(0.f, 0.f, 0.f, 0.f);
    if (ok) {
      const size_t base = transKV ? ((size_t)j * SS + b) : ((size_t)b * SS + j);
      kv = *(const float4*)(qkv + base * 1536 + 512 + h * 64 + tx * 4);
      vv = *(const float4*)(qkv + base * 1536 + 1024 + h * 64 + tx * 4);
    }
    ks[row][tx * 4 + 0] = kv.x;
    ks[row][tx * 4 + 1] = kv.y;
    ks[row][tx * 4 + 2] = kv.z;
    ks[row][tx * 4 + 3] = kv.w;
    vs[row][tx * 4 + 0] = vv.x;
    vs[row][tx * 4 + 1] = vv.y;
    vs[row][tx * 4 + 2] = vv.z;
    vs[row][tx * 4 + 3] = vv.w;
  }
  __syncthreads();

  // Scores: 32x96 = 12 tiles of 16x16, 3 tiles per wave; K = 64.
#pragma unroll
  for (int i = 0; i < 3; ++i) {
    const int tid = wv * 3 + i;
    const int mt = tid / 6;
    const int nt = tid % 6;
    v8f acc = {0.f, 0.f, 0.f, 0.f, 0.f, 0.f, 0.f, 0.f};
#pragma unroll
    for (int kk = 0; kk < 64; kk += 4) {
      v2f a, bfrag;
      a.x = qs[mt * 16 + r][kk + 2 * half];
      a.y = qs[mt * 16 + r][kk + 2 * half + 1];
      bfrag.x = ks[nt * 16 + r][kk + 2 * half];
      bfrag.y = ks[nt * 16 + r][kk + 2 * half + 1];
      acc = wmma4(a, bfrag, acc);
    }
#pragma unroll
    for (int e = 0; e < 8; ++e)
      ps[mt * 16 + e + 8 * half][nt * 16 + r] = acc[e] * 0.125f;  // 1/sqrt(64)
  }
  __syncthreads();

  // Masked softmax over each of the 32 rows (band |rel|<=32, k_abs in [0,S)).
  if (t < 32) {
    const int q = t;
    float mx = -1e30f;
    for (int kk = 0; kk < 96; ++kk) {
      const int rel = kk - 32 - q;
      const int kabs = (n - 1) * 32 + kk;
      const bool valid =
          (rel >= -32) && (rel <= 32) && (kabs >= 0) && (kabs < SS);
      const float sv = valid ? ps[q][kk] : -1e30f;
      ps[q][kk] = sv;
      mx = fmaxf(mx, sv);
    }
    float sum = 0.f;
    for (int kk = 0; kk < 96; ++kk) {
      const float e = __expf(ps[q][kk] - mx);  // masked -> exp(-huge) == 0
      ps[q][kk] = e;
      sum += e;
    }
    const float inv = 1.0f / sum;
    for (int kk = 0; kk < 96; ++kk) ps[q][kk] *= inv;
  }
  __syncthreads();

  // Out: P(32x96) @ V(96x64) = 8 tiles of 16x16, 2 per wave; K = 96.
#pragma unroll
  for (int i = 0; i < 2; ++i) {
    const int tid = wv * 2 + i;
    const int mt = tid >> 2;
    const int nt = tid & 3;
    v8f acc = {0.f, 0.f, 0.f, 0.f, 0.f, 0.f, 0.f, 0.f};
#pragma unroll
    for (int kk = 0; kk < 96; kk += 4) {
      v2f a, bfrag;
      a.x = ps[mt * 16 + r][kk + 2 * half];
      a.y = ps[mt * 16 + r][kk + 2 * half + 1];
      bfrag.x = vs[kk + 2 * half][nt * 16 + r];
      bfrag.y = vs[kk + 2 * half + 1][nt * 16 + r];
      acc = wmma4(a, bfrag, acc);
    }
#pragma unroll
    for (int e = 0; e < 8; ++e) {
      const int s = n * 32 + mt * 16 + e + 8 * half;
      ctx[((size_t)b * SS + s) * 512 + h * 64 + nt * 16 + r] = acc[e];
    }
  }
}

// ---------------------------------------------------------------------------
// Orchestration: v-branch -> h-branch (QKV scratch reused) -> MLP.
// Workspace (floats): qkv 36864*1536 | ctx 36864*512 | comb 36864*1024 |
//                     hid 36864*512   (~528 MB total)
// ---------------------------------------------------------------------------
extern "C" void kernel_launch(void* const* d_in, const int* in_sizes, int n_in,
                              void* d_out, int out_size, void* d_ws,
                              size_t ws_size, hipStream_t stream) {
  (void)in_sizes; (void)n_in; (void)out_size; (void)ws_size;
  const float* x       = (const float*)d_in[0];
  const float* h_in_w  = (const float*)d_in[1];
  const float* h_in_b  = (const float*)d_in[2];
  const float* h_out_w = (const float*)d_in[3];
  const float* h_out_b = (const float*)d_in[4];
  const float* v_in_w  = (const float*)d_in[5];
  const float* v_in_b  = (const float*)d_in[6];
  const float* v_out_w = (const float*)d_in[7];
  const float* v_out_b = (const float*)d_in[8];
  const float* mlp_w1  = (const float*)d_in[9];
  const float* mlp_b1  = (const float*)d_in[10];
  const float* mlp_w2  = (const float*)d_in[11];
  const float* mlp_b2  = (const float*)d_in[12];
  float* out = (float*)d_out;

  float* ws   = (float*)d_ws;
  float* qkv  = ws;                          // MM*1536
  float* ctx  = qkv + (size_t)MM * 1536;     // MM*512
  float* comb = ctx + (size_t)MM * 512;      // MM*1024
  float* hid  = comb + (size_t)MM * 1024;    // MM*512

  const dim3 blk(256);
  const dim3 gQKV(1536 / 128, MM / 128);
  const dim3 gE(512 / 128, MM / 128);
  const dim3 gAttn(HH, NCHUNK, BB);
  const dim3 aBlk(128);

  // Vertical branch
  gemm_wmma_f32<<<gQKV, blk, 0, stream>>>(x, EE, v_in_w, v_in_b, qkv, 1536, 0,
                                          EE, 0);
  attn_wmma_f32<<<gAttn, aBlk, 0, stream>>>(qkv, ctx, 0);
  gemm_wmma_f32<<<gE, blk, 0, stream>>>(ctx, EE, v_out_w, v_out_b, comb, 1024,
                                        512, EE, 0);
  // Horizontal branch (reuses qkv/ctx scratch)
  gemm_wmma_f32<<<gQKV, blk, 0, stream>>>(x, EE, h_in_w, h_in_b, qkv, 1536, 0,
                                          EE, 0);
  attn_wmma_f32<<<gAttn, aBlk, 0, stream>>>(qkv, ctx, 1);
  gemm_wmma_f32<<<gE, blk, 0, stream>>>(ctx, EE, h_out_w, h_out_b, comb, 1024,
                                        0, EE, 0);
  // MLP
  gemm_wmma_f32<<<gE, blk, 0, stream>>>(comb, 1024, mlp_w1, mlp_b1, hid, 512,
                                        0, 1024, 1);
  gemm_wmma_f32<<<gE, blk, 0, stream>>>(hid, EE, mlp_w2, mlp_b2, out, 512, 0,
                                        EE, 0);
}